// TripletLoss_7490422964906
// MI455X (gfx1250) — compile-verified
//
#include <hip/hip_runtime.h>

typedef __attribute__((ext_vector_type(2))) float v2f;
typedef __attribute__((ext_vector_type(8))) float v8f;

#define BATCH    4096
#define DIM      512
#define NEXP     8
#define BSTRIDE  516            // padded LDS row stride: conflict-free, 16B-aligned
#define GROUP    64             // columns staged per LDS group = 4 WMMA tiles
#define SLICE    512            // columns mined per block (8 groups)
#define MARGIN_F 0.5f

// ---------------------------------------------------------------------------
// Kernel 1: per-row squared norm + dominant expert; init per-row mining state.
// ---------------------------------------------------------------------------
__global__ __launch_bounds__(256) void tl_prep_kernel(
    const float* __restrict__ emb, const float* __restrict__ ew,
    float* __restrict__ sq, int* __restrict__ dom,
    unsigned* __restrict__ hpbits, unsigned* __restrict__ hnbits,
    unsigned* __restrict__ flags)
{
    const int gid = blockIdx.x * 256 + threadIdx.x;
    if (gid < BATCH) {
        hpbits[gid] = 0u;                       // max over positives, init 0
        hnbits[gid] = __float_as_uint(1e9f);    // min over negatives, init 1e9
        flags[gid]  = 0u;
    }

    const int lane = threadIdx.x & 31;
    const int wave = threadIdx.x >> 5;
    const int row  = blockIdx.x * 8 + wave;
    if (row >= BATCH) return;

    const float* er = emb + (size_t)row * DIM;
    float s = 0.0f;
    for (int i = lane * 4; i < DIM; i += 128) {
        float4 v = *(const float4*)(er + i);
        s += v.x * v.x + v.y * v.y + v.z * v.z + v.w * v.w;
    }
    #pragma unroll
    for (int off = 16; off > 0; off >>= 1) s += __shfl_xor(s, off, 32);

    if (lane == 0) {
        sq[row] = s;
        const float* w = ew + (size_t)row * NEXP;
        int best = 0; float bv = w[0];
        #pragma unroll
        for (int j = 1; j < NEXP; ++j) {
            float v = w[j];
            if (v > bv) { bv = v; best = j; }   // first-max tie-break == argmax
        }
        dom[row] = best;
    }
}

// ---------------------------------------------------------------------------
// Kernel 2: fused Gram (f32 WMMA, 4-tile register blocking) -> distance ->
// batch-hard mining over a 128-row x 512-col slice per block.
// Grid: (BATCH/128 row blocks) x (BATCH/SLICE column slices).
// ---------------------------------------------------------------------------
__global__ __launch_bounds__(256) void tl_main_kernel(
    const float* __restrict__ emb, const float* __restrict__ sq,
    const int* __restrict__ dom,
    unsigned* __restrict__ hpbits, unsigned* __restrict__ hnbits,
    unsigned* __restrict__ flags)
{
    __shared__ float ldsB[GROUP * BSTRIDE];     // 64 cols x 516 floats = 132 KB

    const int tid  = threadIdx.x;
    const int lane = tid & 31;
    const int wave = tid >> 5;
    const int half = lane >> 4;                 // 0: lanes 0-15, 1: lanes 16-31
    const int l16  = lane & 15;
    const int rowBase   = blockIdx.x * 128 + wave * 16;
    const int sliceBase = blockIdx.y * SLICE;

    // Row metadata (C layout: VGPR v holds row v + 8*half, col = lane&15).
    float srow[8]; int drow[8];
    #pragma unroll
    for (int v = 0; v < 8; ++v) {
        int r = rowBase + v + 8 * half;
        srow[v] = sq[r];
        drow[v] = dom[r];
    }

    float hp[8], hn[8]; int anyp[8], anyn[8];
    #pragma unroll
    for (int v = 0; v < 8; ++v) { hp[v] = 0.0f; hn[v] = 1e9f; anyp[v] = 0; anyn[v] = 0; }

    // A fragment (16x4 f32): lane reads row (rowBase + l16), cols k + 2*half, +1.
    const float* arow = emb + (size_t)(rowBase + l16) * DIM + 2 * half;

    for (int g = 0; g < SLICE / GROUP; ++g) {
        const int groupBase = sliceBase + g * GROUP;

        // Async stage of the 64x512 column block straight into LDS (ASYNCcnt).
        for (int i = tid; i < GROUP * (DIM / 4); i += 256) {
            int r = i >> 7;                     // DIM/4 = 128 float4 per row
            int c = (i & 127) << 2;
            const float* gsrc = emb + (size_t)(groupBase + r) * DIM + c;
            unsigned ldst = (unsigned)(size_t)(&ldsB[r * BSTRIDE + c]);
            asm volatile("global_load_async_to_lds_b128 %0, %1, off"
                         :: "v"(ldst), "v"(gsrc) : "memory");
        }
        asm volatile("s_wait_asynccnt 0x0" ::: "memory");
        __syncthreads();

        // 4 x (16x16) Gram tiles; each A fragment feeds 4 WMMAs.
        v8f cacc[4];
        #pragma unroll
        for (int t = 0; t < 4; ++t) cacc[t] = (v8f){0,0,0,0,0,0,0,0};

        const float* brow0 = ldsB + l16 * BSTRIDE + 2 * half;
        #pragma unroll 4
        for (int k = 0; k < DIM; k += 4) {
            v2f a = *(const v2f*)(arow + k);
            #pragma unroll
            for (int t = 0; t < 4; ++t) {
                v2f b = *(const v2f*)(brow0 + t * (16 * BSTRIDE) + k);
                cacc[t] = __builtin_amdgcn_wmma_f32_16x16x4_f32(
                    false, a, false, b, (short)0, cacc[t], false, false);
            }
        }

        // Fused epilogue: distance + batch-hard masking for the 4 tiles.
        #pragma unroll
        for (int t = 0; t < 4; ++t) {
            const int   colBase = groupBase + t * 16;
            const float sqc  = sq[colBase + l16];
            const int   domc = dom[colBase + l16];
            const int   gcol = colBase + l16;
            #pragma unroll
            for (int v = 0; v < 8; ++v) {
                int   grow = rowBase + v + 8 * half;
                float d2   = srow[v] + sqc - 2.0f * cacc[t][v];
                float dist = __builtin_amdgcn_sqrtf(fmaxf(d2, 1e-12f));
                if (drow[v] == domc) {
                    if (grow != gcol) { hp[v] = fmaxf(hp[v], dist); anyp[v] = 1; }
                } else {                        // different expert => never diagonal
                    hn[v] = fminf(hn[v], dist); anyn[v] = 1;
                }
            }
        }
        __syncthreads();                        // LDS consumed; next group may overwrite
    }

    // Fold the 16 columns held across lanes of each half (masks 1..8 stay in-half).
    #pragma unroll
    for (int v = 0; v < 8; ++v) {
        #pragma unroll
        for (int off = 1; off < 16; off <<= 1) {
            hp[v]   = fmaxf(hp[v], __shfl_xor(hp[v], off, 32));
            hn[v]   = fminf(hn[v], __shfl_xor(hn[v], off, 32));
            anyp[v] |= __shfl_xor(anyp[v], off, 32);
            anyn[v] |= __shfl_xor(anyn[v], off, 32);
        }
    }

    // Merge this slice's per-row results (distances >= 0 -> uint order == float order).
    if (l16 == 0) {
        #pragma unroll
        for (int v = 0; v < 8; ++v) {
            int r = rowBase + v + 8 * half;
            if (anyp[v]) atomicMax(&hpbits[r], __float_as_uint(hp[v]));
            if (anyn[v]) atomicMin(&hnbits[r], __float_as_uint(hn[v]));
            unsigned f = (anyp[v] ? 1u : 0u) | (anyn[v] ? 2u : 0u);
            if (f) atomicOr(&flags[r], f);
        }
    }
}

// ---------------------------------------------------------------------------
// Kernel 3: per-row loss, masked mean -> scalar. Single block, deterministic.
// ---------------------------------------------------------------------------
__global__ __launch_bounds__(256) void tl_reduce_kernel(
    const unsigned* __restrict__ hpbits, const unsigned* __restrict__ hnbits,
    const unsigned* __restrict__ flags, float* __restrict__ out)
{
    __shared__ float ssum[8], scnt[8];
    const int lane = threadIdx.x & 31;
    const int wave = threadIdx.x >> 5;

    float s = 0.0f, c = 0.0f;
    for (int r = threadIdx.x; r < BATCH; r += 256) {
        if (flags[r] == 3u) {                   // has >=1 positive and >=1 negative
            float hpv = __uint_as_float(hpbits[r]);
            float hnv = __uint_as_float(hnbits[r]);
            s += fmaxf(hpv - hnv + MARGIN_F, 0.0f);
            c += 1.0f;
        }
    }
    #pragma unroll
    for (int off = 16; off > 0; off >>= 1) {
        s += __shfl_xor(s, off, 32);
        c += __shfl_xor(c, off, 32);
    }
    if (lane == 0) { ssum[wave] = s; scnt[wave] = c; }
    __syncthreads();
    if (threadIdx.x == 0) {
        float S = 0.0f, C = 0.0f;
        #pragma unroll
        for (int w = 0; w < 8; ++w) { S += ssum[w]; C += scnt[w]; }
        out[0] = S / fmaxf(C, 1.0f);
    }
}

// ---------------------------------------------------------------------------
extern "C" void kernel_launch(void* const* d_in, const int* in_sizes, int n_in,
                              void* d_out, int out_size, void* d_ws, size_t ws_size,
                              hipStream_t stream)
{
    (void)in_sizes; (void)n_in; (void)out_size; (void)ws_size;
    const float* emb = (const float*)d_in[0];   // [4096, 512] fp32
    const float* ew  = (const float*)d_in[1];   // [4096, 8]   fp32
    float*       out = (float*)d_out;           // scalar fp32

    // Workspace (80 KB): sq f32[4096] | dom i32[4096] | hp u32[4096] |
    //                    hn u32[4096] | flags u32[4096]
    float*    sq     = (float*)d_ws;
    int*      dom    = (int*)((char*)d_ws + 4096 * 4);
    unsigned* hpbits = (unsigned*)((char*)d_ws + 4096 * 8);
    unsigned* hnbits = (unsigned*)((char*)d_ws + 4096 * 12);
    unsigned* flags  = (unsigned*)((char*)d_ws + 4096 * 16);

    tl_prep_kernel<<<BATCH / 8, 256, 0, stream>>>(emb, ew, sq, dom, hpbits, hnbits, flags);
    tl_main_kernel<<<dim3(BATCH / 128, BATCH / SLICE), 256, 0, stream>>>(
        emb, sq, dom, hpbits, hnbits, flags);
    tl_reduce_kernel<<<1, 256, 0, stream>>>(hpbits, hnbits, flags, out);
}